// AdaptivePiecewiseLinear_22351009808398
// MI455X (gfx1250) — compile-verified
//
#include <hip/hip_runtime.h>

typedef float v2f __attribute__((ext_vector_type(2)));
typedef float v8f __attribute__((ext_vector_type(8)));

#define BATCH_N   4096
#define NIN       64
#define NOUT      256
#define NPTS      64
#define MTILE     64      // batch rows per workgroup (4 waves x 16)
#define NTILE     64      // output cols per workgroup (4 WMMA n-tiles)
#define LDSV      68      // 64 + 4 pad floats: b64 fragment reads hit 64 distinct banks

__global__ __launch_bounds__(128)
void apl_wmma_kernel(const float* __restrict__ x,
                     const float* __restrict__ positions,
                     const float* __restrict__ values,
                     float* __restrict__ out)
{
    // Double-buffered V slice: [buf][out_col][point(padded)]
    __shared__ __align__(16) float Vbuf[2][NTILE * LDSV];

    const int tid  = threadIdx.x;
    const int wave = tid >> 5;
    const int lane = tid & 31;
    const int l15  = lane & 15;
    const int hi   = lane >> 4;              // 0: lanes 0-15, 1: lanes 16-31
    const int b0   = blockIdx.x * MTILE;
    const int o0   = blockIdx.y * NTILE;
    const int myrow = wave * 16 + l15;       // A-row this lane feeds (dup for hi=0/1)

    // ---- async stage of values[i, o0:o0+64, 0:64] into Vbuf[buf] ----
    auto issue_stage = [&](int i, int buf) {
        const float* vsrc = values + ((size_t)i * NOUT + o0) * NPTS;
        unsigned lbase = (unsigned)(uintptr_t)&Vbuf[buf][0];
        #pragma unroll
        for (int j = 0; j < 8; ++j) {
            int lin = tid + j * 128;              // 0..1023 float4 slots
            int o   = lin >> 4;
            int p4  = (lin & 15) << 2;
            unsigned loff = lbase + (unsigned)(o * LDSV + p4) * 4u;
            unsigned long long gaddr =
                (unsigned long long)(uintptr_t)(vsrc + o * NPTS + p4);
            asm volatile("global_load_async_to_lds_b128 %0, %1, off"
                         :: "v"(loff), "v"(gaddr) : "memory");
        }
    };

    // ---- searchsorted(side='right') on positions[i,0,:]; returns seg+t or -inf ----
    auto search = [&](int i) -> float {
        const float* pos = positions + (size_t)i * (NOUT * NPTS);
        float xv = x[(size_t)(b0 + myrow) * NIN + i];
        int lo = 0, hiN = NPTS;
        #pragma unroll
        for (int s = 0; s < 6; ++s) {
            int mid = (lo + hiN) >> 1;
            if (pos[mid] <= xv) lo = mid + 1; else hiN = mid;
        }
        int seg = lo - 1;
        seg = seg < 0 ? 0 : (seg > NPTS - 2 ? NPTS - 2 : seg);
        float p0v = pos[seg], p1v = pos[seg + 1];
        float t = (xv - p0v) / (p1v - p0v);
        bool in_range = (xv >= pos[0]) && (xv < pos[NPTS - 1]);
        return in_range ? ((float)seg + t) : -1.0e30f;
    };

    v8f acc[4] = {};                         // four 16x16 fp32 C/D tiles per wave

    // prologue: stage V_0, search i=0
    issue_stage(0, 0);
    float s_cur = search(0);
    asm volatile("s_wait_asynccnt 0x0" ::: "memory");
    __syncthreads();

    for (int i = 0; i < NIN; ++i) {
        const int cur = i & 1;
        // fire-and-forget: stage next V slice into the other buffer
        if (i + 1 < NIN) issue_stage(i + 1, cur ^ 1);
        // pipelined search for next i (latency hides under WMMA below)
        float s_nxt = (i + 1 < NIN) ? search(i + 1) : 0.0f;

        // build A fragments in registers via hat weights:
        //   w(k) = max(0, 1 - |k - (seg+t)|)  ->  (1-t) at seg, t at seg+1, else 0
        // A frag layout: lanes 0-15 -> M=lane, K=k0+{0,1}; lanes 16-31 -> M=lane-16, K=k0+{2,3}
        v2f areg[16];
        #pragma unroll
        for (int j = 0; j < 16; ++j) {
            float k0f = (float)(4 * j + 2 * hi);
            areg[j].x = fmaxf(1.0f - fabsf(k0f - s_cur), 0.0f);
            areg[j].y = fmaxf(1.0f - fabsf(k0f + 1.0f - s_cur), 0.0f);
        }

        // B frag layout mirrors A: lanes 0-15 -> N=lane, K=k0+{0,1}; 16-31 -> K=k0+{2,3}
        // Fully unrolled: areg indices are constants -> no movrels/m0 indexing.
        const float* vb = &Vbuf[cur][0];
        #pragma unroll
        for (int kj = 0; kj < 16; ++kj) {
            const int kk = 4 * kj + (hi << 1);
            #pragma unroll
            for (int nt = 0; nt < 4; ++nt) {
                v2f b = *(const v2f*)&vb[(nt * 16 + l15) * LDSV + kk];
                acc[nt] = __builtin_amdgcn_wmma_f32_16x16x4_f32(
                    false, areg[kj], false, b, (short)0, acc[nt], false, false);
            }
        }

        s_cur = s_nxt;
        // publish next V buffer: wait own async loads, then barrier
        asm volatile("s_wait_asynccnt 0x0" ::: "memory");
        __syncthreads();
    }

    // ---- epilogue: C/D layout -> VGPR r: lanes 0-15 M=r, lanes 16-31 M=r+8 ----
    const int row_base = b0 + wave * 16 + (hi << 3);
    #pragma unroll
    for (int nt = 0; nt < 4; ++nt) {
        const int col = o0 + nt * 16 + l15;
        #pragma unroll
        for (int r = 0; r < 8; ++r) {
            out[(size_t)(row_base + r) * NOUT + col] = acc[nt][r];
        }
    }
}

extern "C" void kernel_launch(void* const* d_in, const int* in_sizes, int n_in,
                              void* d_out, int out_size, void* d_ws, size_t ws_size,
                              hipStream_t stream) {
    const float* x         = (const float*)d_in[0];
    const float* positions = (const float*)d_in[1];
    const float* values    = (const float*)d_in[2];
    float* out             = (float*)d_out;
    (void)in_sizes; (void)n_in; (void)out_size; (void)d_ws; (void)ws_size;

    dim3 grid(BATCH_N / MTILE, NOUT / NTILE);   // (64, 4)
    apl_wmma_kernel<<<grid, 128, 0, stream>>>(x, positions, values, out);
}